// UCBBottleneck_56410100465709
// MI455X (gfx1250) — compile-verified
//
#include <hip/hip_runtime.h>
#include <hip/hip_bf16.h>
#include <math.h>

// Problem sizes (fixed by the reference)
#define MROWS 16384   // B*S = 8*2048
#define KIN   512
#define LDIM  256
#define TDIM  1024

typedef __attribute__((ext_vector_type(8))) float v8f;
typedef __attribute__((ext_vector_type(2))) float v2f;

// 16-byte vector type matching the async-LDS builtin's pointee type
typedef int b128_t __attribute__((vector_size(16)));
typedef __attribute__((address_space(1))) b128_t* gptr128_t;
typedef __attribute__((address_space(3))) b128_t* lptr128_t;

#if __has_builtin(__builtin_amdgcn_global_load_async_to_lds_b128)
#define USE_ASYNC_STAGE 1
#else
#define USE_ASYNC_STAGE 0
#endif

#if USE_ASYNC_STAGE
__device__ __forceinline__ void wait_async0() {
#if __has_builtin(__builtin_amdgcn_s_wait_asynccnt)
    __builtin_amdgcn_s_wait_asynccnt(0);
#else
    asm volatile("s_wait_asynccnt 0x0" ::: "memory");
#endif
}
#endif

// ---------------------------------------------------------------------------
// Kernel 1: fused  h = x*w_proj^T + b  ;  logits = h*w_code^T
// One block = one 16-row M tile. 512 threads = 16 waves (wave32).
// A/B/C striping follows CDNA5 ISA f32 16x16x4 layouts:
//   A 16x4:  lanes0-15 -> {K=0,K=1}, lanes16-31 -> {K=2,K=3}, M = lane&15
//   B 4x16:  mirrored: lanes0-15 -> {K=0,K=1}, lanes16-31 -> {K=2,K=3}, N = lane&15
//   C 16x16: vgpr i: lanes0-15 M=i, lanes16-31 M=i+8, N = lane&15
// x tile is staged with GLOBAL_LOAD_ASYNC_TO_LDS_B128 (ASYNCcnt) when the
// toolchain exposes the builtin; otherwise a plain float4 copy.
// ---------------------------------------------------------------------------
__global__ __launch_bounds__(512)
void fused_proj_logits_kernel(const float* __restrict__ x,
                              const float* __restrict__ w_proj,
                              const float* __restrict__ b_proj,
                              const float* __restrict__ w_code,
                              float* __restrict__ logits_ws) {
    __shared__ float sx[16 * KIN];    // 32 KB: x tile  [16][512]
    __shared__ float sh[16 * LDIM];   // 16 KB: h tile  [16][256]

    const int tid  = threadIdx.x;
    const int lane = tid & 31;
    const int wid  = tid >> 5;        // 0..15
    const int hh   = lane >> 4;       // half-wave: 0 or 1
    const int l16  = lane & 15;
    const int m0   = blockIdx.x * 16;

    // ---- stage x tile (contiguous 32 KB) into LDS ----
#if USE_ASYNC_STAGE
    {
        const char* gsrc  = (const char*)(x + (size_t)m0 * KIN);
        const char* lbase = (const char*)sx;
        #pragma unroll
        for (int i = 0; i < 4; ++i) {
            const size_t off = (size_t)tid * 16 + (size_t)i * 8192;
            __builtin_amdgcn_global_load_async_to_lds_b128(
                (gptr128_t)(uintptr_t)(gsrc + off),
                (lptr128_t)(uintptr_t)(lbase + off),
                0, 0);
        }
        wait_async0();   // data resident in LDS for this wave's requests
    }
#else
    {
        const float4* xg  = (const float4*)(x + (size_t)m0 * KIN);
        float4*       sx4 = (float4*)sx;
        #pragma unroll
        for (int i = 0; i < 4; ++i)
            sx4[tid + i * 512] = xg[tid + i * 512];
    }
#endif
    __syncthreads();

    // ---- GEMM1: each wave owns one 16-wide N tile of h (16 waves * 16 = 256)
    {
        const int n0 = wid * 16;
        v8f acc = {};
        const float* wp = w_proj + (size_t)(n0 + l16) * KIN;  // row n, B[k][n]=w_proj[n][k]
        for (int k = 0; k < KIN; k += 4) {
            v2f a, b;
            a.x = sx[l16 * KIN + k + 2 * hh];
            a.y = sx[l16 * KIN + k + 2 * hh + 1];
            b.x = wp[k + 2 * hh];
            b.y = wp[k + 2 * hh + 1];
            acc = __builtin_amdgcn_wmma_f32_16x16x4_f32(
                false, a, false, b, (short)0, acc, false, false);
        }
        const float bb = b_proj[n0 + l16];
        #pragma unroll
        for (int i = 0; i < 8; ++i) {
            const int row = i + hh * 8;
            sh[row * LDIM + n0 + l16] = acc[i] + bb;
        }
    }
    __syncthreads();

    // ---- GEMM2: each wave owns 4 N tiles of logits (16 waves * 4 * 16 = 1024)
    for (int tt = 0; tt < 4; ++tt) {
        const int n0 = (wid * 4 + tt) * 16;
        v8f acc = {};
        const float* wc = w_code + (size_t)(n0 + l16) * LDIM;  // B[k][n]=w_code[n][k]
        for (int k = 0; k < LDIM; k += 4) {
            v2f a, b;
            a.x = sh[l16 * LDIM + k + 2 * hh];
            a.y = sh[l16 * LDIM + k + 2 * hh + 1];
            b.x = wc[k + 2 * hh];
            b.y = wc[k + 2 * hh + 1];
            acc = __builtin_amdgcn_wmma_f32_16x16x4_f32(
                false, a, false, b, (short)0, acc, false, false);
        }
        #pragma unroll
        for (int i = 0; i < 8; ++i) {
            const int row = i + hh * 8;
            logits_ws[(size_t)(m0 + row) * TDIM + n0 + l16] = acc[i];
        }
    }
}

// ---------------------------------------------------------------------------
// Kernel 2: in-place row softmax over T=1024 (TEMP = 1.0). One block per row.
// ---------------------------------------------------------------------------
__global__ __launch_bounds__(256)
void softmax_rows_kernel(float* __restrict__ sc) {
    float* p = sc + (size_t)blockIdx.x * TDIM;
    const int tid  = threadIdx.x;
    const int lane = tid & 31;
    const int wid  = tid >> 5;
    __shared__ float sred[8];

    float4 d = ((const float4*)p)[tid];

    float m = fmaxf(fmaxf(d.x, d.y), fmaxf(d.z, d.w));
    for (int off = 16; off > 0; off >>= 1)
        m = fmaxf(m, __shfl_xor(m, off, 32));
    if (lane == 0) sred[wid] = m;
    __syncthreads();
    float rm = fmaxf(fmaxf(fmaxf(sred[0], sred[1]), fmaxf(sred[2], sred[3])),
                     fmaxf(fmaxf(sred[4], sred[5]), fmaxf(sred[6], sred[7])));
    __syncthreads();

    d.x = __expf(d.x - rm);
    d.y = __expf(d.y - rm);
    d.z = __expf(d.z - rm);
    d.w = __expf(d.w - rm);

    float s = d.x + d.y + d.z + d.w;
    for (int off = 16; off > 0; off >>= 1)
        s += __shfl_xor(s, off, 32);
    if (lane == 0) sred[wid] = s;
    __syncthreads();
    const float rs  = sred[0] + sred[1] + sred[2] + sred[3] +
                      sred[4] + sred[5] + sred[6] + sred[7];
    const float inv = 1.0f / rs;

    d.x *= inv; d.y *= inv; d.z *= inv; d.w *= inv;
    ((float4*)p)[tid] = d;
}

// ---------------------------------------------------------------------------
// Kernel 3: sequential UCB bandit. ONE block, 1024 threads (one lane per arm).
// Per step: bonus_t = sqrt(0.02*(log(csum)-log(c_t))); winner = argmax(s+bonus)
// (ties -> lowest index, matching jnp.argmax); counts[winner] += 1.
// Winner stream IS the final `indices` output (see analysis).
// ---------------------------------------------------------------------------
__global__ __launch_bounds__(1024)
void bandit_kernel(const float* __restrict__ scores,
                   const float* __restrict__ counts,
                   int* __restrict__ idx_ws,
                   float* __restrict__ idx_out) {
    const int t    = threadIdx.x;      // arm id 0..1023
    const int lane = t & 31;
    const int wid  = t >> 5;

    __shared__ float wmaxv[32];
    __shared__ int   wmaxa[32];
    __shared__ float ssum[32];
    __shared__ int   swin;

    float c    = counts[t];
    float logc = __logf(c);

    // block-wide sum of initial counts -> csum (replicated per thread)
    float cs = c;
    for (int off = 16; off > 0; off >>= 1)
        cs += __shfl_xor(cs, off, 32);
    if (lane == 0) ssum[wid] = cs;
    __syncthreads();
    float csum = 0.f;
    #pragma unroll
    for (int i = 0; i < 32; ++i) csum += ssum[i];

    for (int pos = 0; pos < MROWS; ++pos) {
        // prefetch a future row into caches to hide latency of the serial chain
        if (pos + 8 < MROWS)
            __builtin_prefetch(scores + (size_t)(pos + 8) * TDIM + t, 0, 0);

        const float bonus = sqrtf(0.02f * (__logf(csum) - logc));
        float v = scores[(size_t)pos * TDIM + t] + bonus;
        int   a = t;

        // intra-wave argmax (tie -> lower index)
        for (int off = 16; off > 0; off >>= 1) {
            const float ov = __shfl_xor(v, off, 32);
            const int   oa = __shfl_xor(a, off, 32);
            if (ov > v || (ov == v && oa < a)) { v = ov; a = oa; }
        }
        if (lane == 0) { wmaxv[wid] = v; wmaxa[wid] = a; }
        __syncthreads();

        if (wid == 0) {
            float v2 = wmaxv[lane];
            int   a2 = wmaxa[lane];
            for (int off = 16; off > 0; off >>= 1) {
                const float ov = __shfl_xor(v2, off, 32);
                const int   oa = __shfl_xor(a2, off, 32);
                if (ov > v2 || (ov == v2 && oa < a2)) { v2 = ov; a2 = oa; }
            }
            if (lane == 0) {
                swin = a2;
                idx_ws[pos]  = a2;
                idx_out[pos] = (float)a2;
            }
        }
        __syncthreads();

        const int win = swin;
        if (t == win) { c += 1.f; logc = __logf(c); }
        csum += 1.f;
    }
}

// ---------------------------------------------------------------------------
// Kernel 4: codes[row] = w_code[idx[row], :]  (straight-through forward)
// ---------------------------------------------------------------------------
__global__ __launch_bounds__(64)
void gather_codes_kernel(const float* __restrict__ w_code,
                         const int* __restrict__ idx_ws,
                         float* __restrict__ out_codes) {
    const int row = blockIdx.x;
    const int id  = idx_ws[row];
    const float4* src = (const float4*)(w_code + (size_t)id * LDIM);
    float4*       dst = (float4*)(out_codes + (size_t)row * LDIM);
    dst[threadIdx.x] = src[threadIdx.x];
}

// ---------------------------------------------------------------------------
extern "C" void kernel_launch(void* const* d_in, const int* in_sizes, int n_in,
                              void* d_out, int out_size, void* d_ws, size_t ws_size,
                              hipStream_t stream) {
    const float* x      = (const float*)d_in[0];
    const float* w_proj = (const float*)d_in[1];
    const float* b_proj = (const float*)d_in[2];
    const float* w_code = (const float*)d_in[3];
    const float* counts = (const float*)d_in[4];

    float* out_codes = (float*)d_out;                           // 16384*256 floats
    float* out_idx   = out_codes + (size_t)MROWS * LDIM;        // 16384 floats

    float* ws_scores = (float*)d_ws;                            // 64 MB (L2-resident)
    int*   ws_idx    = (int*)(ws_scores + (size_t)MROWS * TDIM);

    fused_proj_logits_kernel<<<MROWS / 16, 512, 0, stream>>>(
        x, w_proj, b_proj, w_code, ws_scores);
    softmax_rows_kernel<<<MROWS, 256, 0, stream>>>(ws_scores);
    bandit_kernel<<<1, 1024, 0, stream>>>(ws_scores, counts, ws_idx, out_idx);
    gather_codes_kernel<<<MROWS, 64, 0, stream>>>(w_code, ws_idx, out_codes);
}